// GGNN_NoGRU_NoEdgeNets_88252987998922
// MI455X (gfx1250) — compile-verified
//
#include <hip/hip_runtime.h>
#include <math.h>

#define BB 8
#define NN 20000
#define DD 128
#define EE 64000
#define ESETS 3
#define NPASS 5
#define HIDN 80
#define OUTN 10
#define POOL_CHUNKS 64

typedef __attribute__((ext_vector_type(2))) float v2f;
typedef __attribute__((ext_vector_type(8))) float v8f;

// ---------------------------------------------------------------------------
// Pass init: next = cur  (pass result is nodes + sum(messages), so the
// scatter target starts as a copy of the current node state). float4 copies.
// ---------------------------------------------------------------------------
__global__ void copy_f4(const float4* __restrict__ s, float4* __restrict__ d, int n4) {
    int i = blockIdx.x * blockDim.x + threadIdx.x;
    if (i < n4) d[i] = s[i];
}

__global__ void zero_f(float* __restrict__ p, int n) {
    int i = blockIdx.x * blockDim.x + threadIdx.x;
    if (i < n) p[i] = 0.0f;
}

// ---------------------------------------------------------------------------
// Message passing: one wave32 per edge. Lane l moves 4 floats (float4 gather
// from src row -> 4x global_atomic_add_f32 into dst row). Edges layout is
// (B, 3, E, 2) with [dst, src] pairs -> flattened wave index maps directly.
// Working set (82MB nodes + 12MB edges) is L2-resident on MI455X (192MB L2).
// ---------------------------------------------------------------------------
__global__ void edge_scatter(const float* __restrict__ src_nodes,
                             float* __restrict__ dst_nodes,
                             const int* __restrict__ edges,
                             int total_edges) {
    int tid  = blockIdx.x * blockDim.x + threadIdx.x;
    int w    = tid >> 5;       // wave index == flattened (b, set, e)
    int lane = tid & 31;
    if (w >= total_edges) return;
    int b   = w / (ESETS * EE);
    int dst = edges[2 * w + 0];
    int src = edges[2 * w + 1];
    const float4* srow = (const float4*)(src_nodes + ((size_t)b * NN + src) * DD);
    float*        drow = dst_nodes + ((size_t)b * NN + dst) * DD + lane * 4;
    float4 v = srow[lane];
    unsafeAtomicAdd(drow + 0, v.x);
    unsafeAtomicAdd(drow + 1, v.y);
    unsafeAtomicAdd(drow + 2, v.z);
    unsafeAtomicAdd(drow + 3, v.w);
}

// ---------------------------------------------------------------------------
// pooled[b][d] = sum_n nodes[b][n][d]. 128 threads per block own one feature
// each (coalesced across d), POOL_CHUNKS blocks stride over N, atomic combine.
// ---------------------------------------------------------------------------
__global__ void pool_sum(const float* __restrict__ nodes, float* __restrict__ pooled) {
    int b = blockIdx.x / POOL_CHUNKS;
    int c = blockIdx.x % POOL_CHUNKS;
    int d = threadIdx.x;
    const float* base = nodes + (size_t)b * NN * DD + d;
    float acc = 0.0f;
    for (int n = c; n < NN; n += POOL_CHUNKS) acc += base[(size_t)n * DD];
    unsafeAtomicAdd(&pooled[b * DD + d], acc);
}

// ---------------------------------------------------------------------------
// One WMMA GEMM layer on a single wave32. A and W both live in LDS, both
// pre-padded so the k-loop has NO bounds checks: each step is two
// ds_load_b64 + one v_wmma_f32_16x16x4_f32, no EXEC manipulation.
// V_WMMA_F32_16X16X4_F32 layouts (ISA 7.12.2):
//   A: lane = M, vgpr pair = {K=k0+2h, k0+2h+1}, h = lane/16
//   B: lane = N, vgpr pair = {K=k0+2h, k0+2h+1}
//   C/D: element i in lane l -> row = i + 8*(l/16), col = l%16
// ---------------------------------------------------------------------------
__device__ __forceinline__ void mlp_layer(const float* As, int lda, int ksteps,
                                          const float* Ws, int ldw,
                                          const float* __restrict__ bias, int ncol,
                                          float* Os, int ldo, int nrows,
                                          bool leaky, int lane) {
    int h  = lane >> 4;
    int nl = lane & 15;
    int ntiles = (ncol + 15) >> 4;
    for (int nt = 0; nt < ntiles; ++nt) {
        int col = nt * 16 + nl;
        v8f acc = {};
        for (int ks = 0; ks < ksteps; ++ks) {
            int ka = ks * 4 + 2 * h;
            v2f a, bfr;
            a.x   = As[nl * lda + ka];
            a.y   = As[nl * lda + ka + 1];
            bfr.x = Ws[ka * ldw + col];
            bfr.y = Ws[(ka + 1) * ldw + col];
            acc = __builtin_amdgcn_wmma_f32_16x16x4_f32(
                false, a, false, bfr, (short)0, acc, false, false);
        }
        if (col < ncol) {
            float bb = bias[col];
#pragma unroll
            for (int i = 0; i < 8; ++i) {
                int row = i + 8 * h;
                if (row < nrows) {
                    float v = acc[i] + bb;
                    if (leaky) v = (v > 0.0f) ? v : 0.01f * v;
                    Os[row * ldo + col] = v;
                }
            }
        }
    }
}

// ---------------------------------------------------------------------------
// Head: y = relu(nan_to_0(log(pooled))); y[+inf] = row finite max;
// x = [y, problem_type]; 3-layer MLP via f32 WMMA. Single wave32 block.
// Weights staged (zero-padded) into one reused LDS buffer per layer.
// ---------------------------------------------------------------------------
__global__ __launch_bounds__(32) void head_kernel(const float* __restrict__ pooled,
                                                  const float* __restrict__ ptype,
                                                  const float* __restrict__ w1,
                                                  const float* __restrict__ b1,
                                                  const float* __restrict__ w2,
                                                  const float* __restrict__ b2,
                                                  const float* __restrict__ w3,
                                                  const float* __restrict__ b3,
                                                  float* __restrict__ out) {
    __shared__ float Xs[16][132];    // x, K padded 129 -> 132, rows 8..15 zero
    __shared__ float H1[16][HIDN];
    __shared__ float H2[16][HIDN];
    __shared__ float Ws[132 * HIDN]; // reused weight buffer (largest: w1 padded)
    int l = threadIdx.x;

    for (int i = l; i < 16 * 132; i += 32) (&Xs[0][0])[i] = 0.0f;

    // Build y rows: 32 lanes x 4 features per row.
    for (int r = 0; r < BB; ++r) {
        float y[4];
        float m = -INFINITY;
#pragma unroll
        for (int j = 0; j < 4; ++j) {
            float v = pooled[r * DD + l * 4 + j];
            float t = logf(v);
            if (t != t) t = 0.0f;        // nan -> 0
            t = fmaxf(t, 0.0f);          // relu (also kills -inf from log(0))
            y[j] = t;
            if (t < INFINITY) m = fmaxf(m, t);
        }
#pragma unroll
        for (int off = 16; off >= 1; off >>= 1)
            m = fmaxf(m, __shfl_xor(m, off, 32));
#pragma unroll
        for (int j = 0; j < 4; ++j) {
            float t = y[j];
            if (t == INFINITY) t = m;    // +inf -> row finite max
            Xs[r][l * 4 + j] = t;
        }
    }
    if (l < BB) Xs[l][DD] = ptype[l];    // concat problem_type at column 128

    // ---- layer 1: stage w1 as 132x80 (rows 129..131 zero) ----
    for (int i = l; i < 132 * HIDN; i += 32) {
        int k = i / HIDN, c = i % HIDN;
        Ws[i] = (k < DD + 1) ? w1[k * HIDN + c] : 0.0f;
    }
    __syncthreads();
    mlp_layer(&Xs[0][0], 132, 33, Ws, HIDN, b1, HIDN, &H1[0][0], HIDN, 16, true, l);
    __syncthreads();

    // ---- layer 2: stage w2 as 80x80 (exact) ----
    for (int i = l; i < HIDN * HIDN; i += 32) Ws[i] = w2[i];
    __syncthreads();
    mlp_layer(&H1[0][0], HIDN, 20, Ws, HIDN, b2, HIDN, &H2[0][0], HIDN, 16, true, l);
    __syncthreads();

    // ---- layer 3: stage w3 as 80x16 (cols 10..15 zero) ----
    for (int i = l; i < HIDN * 16; i += 32) {
        int k = i / 16, c = i % 16;
        Ws[i] = (c < OUTN) ? w3[k * OUTN + c] : 0.0f;
    }
    __syncthreads();
    mlp_layer(&H2[0][0], HIDN, 20, Ws, 16, b3, OUTN, out, OUTN, BB, false, l);
}

// ---------------------------------------------------------------------------
extern "C" void kernel_launch(void* const* d_in, const int* in_sizes, int n_in,
                              void* d_out, int out_size, void* d_ws, size_t ws_size,
                              hipStream_t stream) {
    const float* nodes = (const float*)d_in[0];
    const float* ptype = (const float*)d_in[1];
    const float* w1    = (const float*)d_in[2];
    const float* b1    = (const float*)d_in[3];
    const float* w2    = (const float*)d_in[4];
    const float* b2    = (const float*)d_in[5];
    const float* w3    = (const float*)d_in[6];
    const float* b3    = (const float*)d_in[7];
    const int*   edges = (const int*)d_in[8];
    float*       out   = (float*)d_out;

    const size_t nelem = (size_t)BB * NN * DD;     // 20,480,000 floats
    float* buf0   = (float*)d_ws;
    float* buf1   = buf0 + nelem;
    float* pooled = buf1 + nelem;                  // BB*DD floats

    const int n4           = (int)(nelem / 4);     // float4 count
    const int copy_blocks  = (n4 + 255) / 256;
    const int total_edges  = BB * ESETS * EE;      // 1,536,000 waves
    const int edge_threads = total_edges * 32;
    const int edge_blocks  = (edge_threads + 255) / 256;

    // 5 sequential message-passing rounds, double-buffered.
    const float* cur = nodes;
    float* nxt = buf0;
    for (int p = 0; p < NPASS; ++p) {
        copy_f4<<<copy_blocks, 256, 0, stream>>>((const float4*)cur, (float4*)nxt, n4);
        edge_scatter<<<edge_blocks, 256, 0, stream>>>(cur, nxt, edges, total_edges);
        cur = nxt;
        nxt = (nxt == buf0) ? buf1 : buf0;
    }
    // cur == buf0 after 5 passes

    zero_f<<<(BB * DD + 255) / 256, 256, 0, stream>>>(pooled, BB * DD);
    pool_sum<<<BB * POOL_CHUNKS, DD, 0, stream>>>(cur, pooled);

    head_kernel<<<1, 32, 0, stream>>>(pooled, ptype, w1, b1, w2, b2, w3, b3, out);
}